// BlockEnd_53266184405691
// MI455X (gfx1250) — compile-verified
//
#include <hip/hip_runtime.h>

// Fused masked-residual-GEMM for MI455X (gfx1250, wave32, WMMA).
// Y[b] = mask_rows( relu( R[b](128x128) @ W(128x128) + N[b] ), atoms[b] )
//
// Memory-bound problem (~805 MB @ 23.3 TB/s => ~35us floor), so we use the
// full-precision V_WMMA_F32_16X16X4_F32 path: matrix throughput is not the
// limiter and f32 keeps bit-accuracy vs the f32 reference.

typedef __attribute__((ext_vector_type(2))) float v2f;
typedef __attribute__((ext_vector_type(8))) float v8f;

#define BDIM 4096
#define ADIM 128
#define FDIM 128
#define RFDIM 128
// LDS row stride (floats). 132 = 128+4 makes both A-frag reads
// (bank = 4*row + {0,1}) and transposed-W B-frag reads (bank = 4*col + k)
// conflict-free across the 32 lanes.
#define LSTR 132

__global__ __launch_bounds__(256)
void molgemm_relu_mask(const float* __restrict__ node,   // (B, A, F)
                       const float* __restrict__ resid,  // (B, A, RF)
                       const float* __restrict__ w,      // (RF, F)
                       const int*   __restrict__ mol,    // (B, 2) int32
                       float*       __restrict__ out)    // (B, A, F)
{
    __shared__ float sWT[FDIM * LSTR];   // W transposed: sWT[col][k]
    __shared__ float sA [ADIM * LSTR];   // residual tile: sA[row][k]

    const int b    = blockIdx.x;
    const int tid  = threadIdx.x;
    const int wave = tid >> 5;          // 0..7 -> M-strip
    const int lane = tid & 31;
    const int hl   = lane & 15;         // half-lane index
    const int hi   = lane >> 4;         // 0: lanes 0-15, 1: lanes 16-31

    // ---- Stage W (transposed) into LDS: coalesced 128B global reads ----
    for (int i = tid; i < RFDIM * FDIM; i += 256) {
        const int k = i >> 7;           // w row  (RF index)
        const int c = i & 127;          // w col  (F index)
        sWT[c * LSTR + k] = w[i];
    }
    // ---- Stage residual[b] into LDS: coalesced 128B global reads ----
    const float* __restrict__ Rb = resid + (size_t)b * (ADIM * RFDIM);
    for (int i = tid; i < ADIM * RFDIM; i += 256) {
        const int r = i >> 7;
        const int k = i & 127;
        sA[r * LSTR + k] = Rb[i];
    }
    __syncthreads();

    const float* __restrict__ Nb = node + (size_t)b * (ADIM * FDIM);
    float*       __restrict__ Ob = out  + (size_t)b * (ADIM * FDIM);

    const int mbase = wave * 16;                // this wave's 16-row strip
    const int crow0 = mbase + (hi ? 8 : 0);     // C/D layout: VGPR i -> M = i (+8 hi half)

    // ---- Init accumulators from node_features (the "+ N" bias) ----
    // C/D 16x16 f32 layout: acc[n][i] = element (M = crow0+i, N = n*16+hl)
    v8f acc[8];
#pragma unroll
    for (int n = 0; n < 8; ++n) {
#pragma unroll
        for (int i = 0; i < 8; ++i) {
            acc[n][i] = Nb[(size_t)(crow0 + i) * FDIM + n * 16 + hl];
        }
    }

    // ---- Main loop: 32 K-steps of V_WMMA_F32_16X16X4_F32 over 8 N-tiles ----
    // A 16x4 layout: lane(0-15)=row M, K = kt*4 + {0,1}; lanes 16-31: K += 2.
    // B 4x16 layout mirrors A with N across lanes (read from transposed W).
    const int arow = mbase + hl;
#pragma unroll 4
    for (int kt = 0; kt < 32; ++kt) {
        const int kb = kt * 4 + hi * 2;
        v2f afrag;
        afrag.x = sA[arow * LSTR + kb];
        afrag.y = sA[arow * LSTR + kb + 1];
#pragma unroll
        for (int n = 0; n < 8; ++n) {
            v2f bfrag;
            const int bcol = n * 16 + hl;
            bfrag.x = sWT[bcol * LSTR + kb];
            bfrag.y = sWT[bcol * LSTR + kb + 1];
            acc[n] = __builtin_amdgcn_wmma_f32_16x16x4_f32(
                /*neg_a=*/false, afrag,
                /*neg_b=*/false, bfrag,
                /*c_mod=*/(short)0, acc[n],
                /*reuse_a=*/false, /*reuse_b=*/false);
        }
    }

    // ---- Epilogue: ReLU + row mask (row < atoms[b]) + coalesced store ----
    const int atoms = mol[b * 2];   // mol_slice[b][0]
#pragma unroll
    for (int n = 0; n < 8; ++n) {
#pragma unroll
        for (int i = 0; i < 8; ++i) {
            const int row = crow0 + i;
            float v = acc[n][i];
            v = v > 0.0f ? v : 0.0f;
            v = (row < atoms) ? v : 0.0f;
            Ob[(size_t)row * FDIM + n * 16 + hl] = v;
        }
    }
}

extern "C" void kernel_launch(void* const* d_in, const int* in_sizes, int n_in,
                              void* d_out, int out_size, void* d_ws, size_t ws_size,
                              hipStream_t stream) {
    const float* node  = (const float*)d_in[0];   // (B, A, F)   f32
    const float* resid = (const float*)d_in[1];   // (B, A, RF)  f32
    const float* w     = (const float*)d_in[2];   // (RF, F)     f32
    const int*   mol   = (const int*)d_in[3];     // (B, 2)      int32
    float*       outp  = (float*)d_out;           // (B, A, F)   f32

    dim3 grid(BDIM);
    dim3 block(256);
    molgemm_relu_mask<<<grid, block, 0, stream>>>(node, resid, w, mol, outp);
}